// Temporal_Operator_36825049596150
// MI455X (gfx1250) — compile-verified
//
#include <hip/hip_runtime.h>

typedef __attribute__((ext_vector_type(2))) float v2f;
typedef __attribute__((ext_vector_type(8))) float v8f;

#define B_DIM 64
#define T_DIM 2048
#define D_DIM 128

// One wave32 computes a 16(t) x 16(d) output tile.
// Window sums are a banded matmul: sum[m,n] = Σ_j band[m,j] * V[T0-15+j, n],
// band[m,j] = 1 iff m <= j <= m+15  (j in 0..31, K padded; j=31 band==0).
// Done with 8 chained V_WMMA_F32_16X16X4_F32 steps (full f32 precision).
__global__ __launch_bounds__(256) void temporal_softmax_window_kernel(
    const float* __restrict__ x, const int* __restrict__ scale_p,
    float* __restrict__ out)
{
    const float s   = (float)(*scale_p);
    const int tid   = threadIdx.x;
    const int lane  = tid & 31;
    const int wave  = tid >> 5;        // 0..7 -> which d-tile
    const int m     = lane & 15;       // A-row (t within tile) == B-col (d within tile)
    const int hi    = lane >> 4;       // lane half selects K pairs {0,1} vs {2,3}

    const int blk = blockIdx.x;        // b * (T/16) + t-tile
    const int b   = blk >> 7;
    const int T0  = (blk & 127) << 4;
    const int d   = (wave << 4) + m;

    const float* xb = x + (size_t)b * (T_DIM * D_DIM);

    v8f accE = {0.f,0.f,0.f,0.f,0.f,0.f,0.f,0.f};   // Σ e^{s z}
    v8f accN = {0.f,0.f,0.f,0.f,0.f,0.f,0.f,0.f};   // Σ z e^{s z}

    #pragma unroll
    for (int step = 0; step < 8; ++step) {
        // This lane holds K rows (kb, kb+1):
        //   A layout: VGPR0 = K{0,2}, VGPR1 = K{1,3} across lane halves
        //   B layout mirrors (rows of B striped across lane halves)
        const int kb = step * 4 + hi * 2;

        // Constant band matrix A[m, j] in {0,1}
        v2f a;
        a.x = (kb     >= m && kb     <= m + 15) ? 1.0f : 0.0f;
        a.y = (kb + 1 >= m && kb + 1 <= m + 15) ? 1.0f : 0.0f;

        // Source times for j = kb, kb+1; clamp left edge (h0 = repeat of x[:,0,:])
        // and clamp the never-used j==31 tap at the right edge (band is 0 there).
        int tx = T0 - 15 + kb;
        int ty = tx + 1;
        tx = min(max(tx, 0), T_DIM - 1);
        ty = min(max(ty, 0), T_DIM - 1);

        const float zx = xb[(size_t)tx * D_DIM + d];
        const float zy = xb[(size_t)ty * D_DIM + d];
        const float ex = __expf(zx * s);
        const float ey = __expf(zy * s);

        v2f bE; bE.x = ex;      bE.y = ey;
        v2f bN; bN.x = zx * ex; bN.y = zy * ey;

        accE = __builtin_amdgcn_wmma_f32_16x16x4_f32(
                   false, a, false, bE, (short)0, accE, false, false);
        accN = __builtin_amdgcn_wmma_f32_16x16x4_f32(
                   false, a, false, bN, (short)0, accN, false, false);
    }

    // C/D layout: VGPR r -> row M = r (lanes 0-15) / M = r+8 (lanes 16-31), col N = lane&15
    float* ob = out + (size_t)b * (T_DIM * D_DIM);
    #pragma unroll
    for (int r = 0; r < 8; ++r) {
        const int t = T0 + r + hi * 8;
        ob[(size_t)t * D_DIM + d] = accN[r] / accE[r];
    }
}

extern "C" void kernel_launch(void* const* d_in, const int* in_sizes, int n_in,
                              void* d_out, int out_size, void* d_ws, size_t ws_size,
                              hipStream_t stream) {
    const float* x     = (const float*)d_in[0];
    const int*   scale = (const int*)d_in[1];
    float*       out   = (float*)d_out;

    dim3 grid(B_DIM * (T_DIM / 16));   // 8192 blocks: (b, t-tile)
    dim3 block(256);                   // 8 waves, one 16x16 d-tile each
    hipLaunchKernelGGL(temporal_softmax_window_kernel, grid, block, 0, stream,
                       x, scale, out);
}